// AntiAliasInterpolation2d_42923903156617
// MI455X (gfx1250) — compile-verified
//
#include <hip/hip_runtime.h>

// ---------------------------------------------------------------------------
// AntiAliasInterpolation2d for MI455X (gfx1250)
//  - folds 13x13 depthwise Gaussian + bilinear x0.25 into a separable
//    14-tap stride-4 decimating filter g[s] = 0.5*(kn[s]+kn[s-1])
//  - async global->LDS staging (B128 on interior tiles), banded WMMA
//    f32 16x16x4 compute fed from an LDS band table (no predication)
// ---------------------------------------------------------------------------

typedef float v2f __attribute__((ext_vector_type(2)));
typedef float v8f __attribute__((ext_vector_type(8)));
typedef int   v4i __attribute__((vector_size(16)));        // matches builtin sig

typedef __attribute__((address_space(1))) int* gas1_i32p;  // global int*
typedef __attribute__((address_space(3))) int* las3_i32p;  // LDS int*
typedef __attribute__((address_space(1))) v4i* gas1_v4ip;  // global v4i*
typedef __attribute__((address_space(3))) v4i* las3_v4ip;  // LDS v4i*

#ifndef __has_builtin
#define __has_builtin(x) 0
#endif

#if __has_builtin(__builtin_amdgcn_global_load_async_to_lds_b32)
#define HAVE_ASYNC_B32 1
#else
#define HAVE_ASYNC_B32 0
#endif

#if __has_builtin(__builtin_amdgcn_global_load_async_to_lds_b128)
#define HAVE_ASYNC_B128 1
#else
#define HAVE_ASYNC_B128 0
#endif

#if __has_builtin(__builtin_amdgcn_s_wait_asynccnt)
#define HAVE_WAIT_ASYNC 1
#else
#define HAVE_WAIT_ASYNC 0
#endif

#define IMG_H 512
#define IMG_W 512
#define OUT_H 128
#define OUT_W 128
#define TILE  16        // output tile is 16x16
#define INW   74        // input window rows: 4*16 - 4 + 14 = 74
#define XSH   3         // col shift: window col c lives at LDS col c+3
#define LDSW  84        // LDS row stride (dwords): 84 mod 64 = 20 -> conflict-free
#define LDSR  80        // padded rows (5 WMMA M-tiles of 16)
#define HBR   84        // HB rows (window row r stored at r+3, max 82)
#define KCH   20        // K chunks of 4 (covers LDS cols / HB rows 0..79)
#define NTHREADS 160    // 5 wave32

__global__ __launch_bounds__(NTHREADS)
void aa_interp_kernel(const float* __restrict__ x,
                      const float* __restrict__ w,
                      float* __restrict__ out) {
  __shared__ float sX[LDSR * LDSW];   // staged input tile (zero-filled halo)
  __shared__ float sHB[HBR * 16];     // horizontally blurred+decimated tile
  __shared__ float sB[LDSR * 16];     // band table: sB[k*16+n] = g[k-3-4n]
  __shared__ float sKN[16];           // normalized 1-D gaussian (row sums)

  const int tid  = threadIdx.x;
  const int lane = tid & 31;
  const int wv   = tid >> 5;          // wave id 0..4
  const int nlo  = lane & 15;         // M (A/C row) or N (B/C col) index
  const int hi   = lane >> 4;         // lane-half selector (K offset +2)

  const int bxc = blockIdx.x & 7;     // output col-block (0..7)
  const int byr = blockIdx.x >> 3;    // output row-block (0..7)
  const int ch  = blockIdx.y;         // channel
  const int img = blockIdx.z;         // batch index

  // ---- Phase 0: 1-D taps + zero LDS -------------------------------------
  // weight is (3,1,13,13), identical channels, sums to 1 => normalized 1-D
  // taps are the row sums of channel 0.
  if (tid < 13) {
    float s = 0.f;
    #pragma unroll
    for (int q = 0; q < 13; ++q) s += w[tid * 13 + q];
    sKN[tid] = s;
  }
  if (tid >= 13 && tid < 16) sKN[tid] = 0.f;
  for (int i = tid; i < LDSR * LDSW; i += NTHREADS) sX[i] = 0.f;
  for (int i = tid; i < HBR * 16;    i += NTHREADS) sHB[i] = 0.f;
  __syncthreads();

  // ---- Phase 0b: band table  sB[k*16+n] = g[k-3-4n],
  //      g[d] = 0.5*(kn[d] + kn[d-1]) for d in [0,13] ----------------------
  for (int i = tid; i < LDSR * 16; i += NTHREADS) {
    const int k = i >> 4, n = i & 15;
    const int d = k - XSH - 4 * n;
    float v = 0.f;
    if ((unsigned)d < 14u) {
      const float a = (d < 13) ? sKN[d] : 0.f;
      const float b = (d > 0) ? sKN[d - 1] : 0.f;
      v = 0.5f * (a + b);
    }
    sB[i] = v;
  }

  // ---- Phase 1: stage input window into LDS (async copies) ---------------
  const int rs  = 4 * (byr * TILE) - 5;       // first input row of window
  const int cs  = 4 * (bxc * TILE) - 5;       // first input col of window
  const int c0a = cs - XSH;                   // 16B-aligned staged col origin
  const float* gbase = x + ((size_t)(img * 3 + ch)) * IMG_H * IMG_W;

  const bool interior = (byr >= 1) && (byr <= 6) && (bxc >= 1) && (bxc <= 6);
  if (interior) {
    // 74 rows x 20 float4 (80 aligned cols): one b128 per lane-iteration
    for (int q = tid; q < INW * (LDSW / 4 - 1); q += NTHREADS) {
      const int tr = q / 20;
      const int i4 = q - tr * 20;
      const float* gp = gbase + (rs + tr) * IMG_W + c0a + 4 * i4;
      float*       lp = &sX[tr * LDSW + 4 * i4];
#if HAVE_ASYNC_B128
      __builtin_amdgcn_global_load_async_to_lds_b128(
          (gas1_v4ip)gp, (las3_v4ip)lp, 0, 0);
#elif HAVE_ASYNC_B32
      #pragma unroll
      for (int e = 0; e < 4; ++e)
        __builtin_amdgcn_global_load_async_to_lds_b32(
            (gas1_i32p)(gp + e), (las3_i32p)(lp + e), 0, 0);
#else
      *(float4*)lp = *(const float4*)gp;
#endif
    }
  } else {
    // boundary: checked scalar copies over pre-zeroed LDS
    for (int idx = tid; idx < INW * INW; idx += NTHREADS) {
      const int tr = idx / INW;
      const int tc = idx - tr * INW;
      const int r = rs + tr, c = cs + tc;
      if ((unsigned)r < (unsigned)IMG_H && (unsigned)c < (unsigned)IMG_W) {
        const float* gp = gbase + r * IMG_W + c;
        float*       lp = &sX[tr * LDSW + tc + XSH];
#if HAVE_ASYNC_B32
        __builtin_amdgcn_global_load_async_to_lds_b32(
            (gas1_i32p)gp, (las3_i32p)lp, 0, 0);
#else
        *lp = __builtin_nontemporal_load(gp);
#endif
      }
    }
  }
#if HAVE_ASYNC_B32 || HAVE_ASYNC_B128
#if HAVE_WAIT_ASYNC
  __builtin_amdgcn_s_wait_asynccnt(0);
#else
  asm volatile("s_wait_asynccnt 0" ::: "memory");
#endif
#endif
  __syncthreads();

  // ---- Phase 2: horizontal decimating blur via banded WMMA ----------------
  // hb[r,n] = sum_k sX[r,k] * sB[k,n]; K = 80 as 20 chunks of 4.
  // Each wave owns one 16-row M-tile.
  {
    const int rb = wv * 16;            // window rows rb..rb+15
    v8f acc = {};
    for (int kb = 0; kb < KCH; ++kb) {
      const int k0 = 4 * kb + 2 * hi;  // VGPR0 K index for this lane-half
      v2f a, b;
      a.x = sX[(rb + nlo) * LDSW + k0];
      a.y = sX[(rb + nlo) * LDSW + k0 + 1];
      b.x = sB[k0 * 16 + nlo];
      b.y = sB[(k0 + 1) * 16 + nlo];
      acc = __builtin_amdgcn_wmma_f32_16x16x4_f32(
          false, a, false, b, (short)0, acc, false, false);
    }
    // C layout: VGPR v -> row v (lanes 0-15) / v+8 (lanes 16-31), col lane%16
    #pragma unroll
    for (int v = 0; v < 8; ++v)
      sHB[(rb + v + 8 * hi + XSH) * 16 + nlo] = acc[v];
  }
  __syncthreads();

  // ---- Phase 3: vertical decimating blur via banded WMMA (wave 0) ---------
  // out[m,n] = sum_k sB[k,m] * sHB[k,n]  (A[m,k] = g[k-3-4m] = sB[k*16+m])
  if (wv == 0) {
    v8f acc = {};
    for (int kb = 0; kb < KCH; ++kb) {
      const int k0 = 4 * kb + 2 * hi;
      v2f a, b;
      a.x = sB[k0 * 16 + nlo];         // M = nlo here
      a.y = sB[(k0 + 1) * 16 + nlo];
      b.x = sHB[k0 * 16 + nlo];
      b.y = sHB[(k0 + 1) * 16 + nlo];
      acc = __builtin_amdgcn_wmma_f32_16x16x4_f32(
          false, a, false, b, (short)0, acc, false, false);
    }
    float* obase = out + ((size_t)(img * 3 + ch)) * OUT_H * OUT_W;
    const int ocol = bxc * TILE + nlo;
    #pragma unroll
    for (int v = 0; v < 8; ++v) {
      const int orow = byr * TILE + v + 8 * hi;
      obase[orow * OUT_W + ocol] = acc[v];
    }
  }
}

extern "C" void kernel_launch(void* const* d_in, const int* in_sizes, int n_in,
                              void* d_out, int out_size, void* d_ws, size_t ws_size,
                              hipStream_t stream) {
  const float* x = (const float*)d_in[0];   // (32,3,512,512) fp32
  const float* w = (const float*)d_in[1];   // (3,1,13,13) fp32
  float* out = (float*)d_out;               // (32,3,128,128) fp32

  dim3 grid(8 * 8, 3, 32);                  // 16x16 output tiles, channel, batch
  aa_interp_kernel<<<grid, NTHREADS, 0, stream>>>(x, w, out);
}